// Talos_17566416241179
// MI455X (gfx1250) — compile-verified
//
#include <hip/hip_runtime.h>
#include <math.h>

// Problem constants from the reference
#define T_LEN   240
#define KLAGS   61
#define HALF_K  30
#define PAD_LEN 304   // 240 + 60 pad, rounded up so lanes 61..63 stay in-bounds
#define S_SUBJ  64
#define ACC_LD  64    // lag dimension padded to 64 (cols 61..63 stay zero)

typedef float v2f __attribute__((ext_vector_type(2)));
typedef float v8f __attribute__((ext_vector_type(8)));

// ---------------------------------------------------------------------------
// Kernel 1: per-row 61-lag correlation + window power sums (memory-bound part)
// One 64-thread block (2 wave32) per row b; lane == lag k.
//   corr[k] = sum_t yh[t] * yp[t+k]
//   s2[k]   = sum_t yp[t+k]^2          (windowed sum of y^2, zeros outside)
//   h       = sum_t yh[t]^2
//   240*mse[b,k] = s2 + h - 2*corr
// Accumulated per subject: acc[s, k] += 240*mse[b,k]  (f32 global atomics,
// 3904 distinct addresses -> negligible contention at L2).
// ---------------------------------------------------------------------------
__global__ __launch_bounds__(64) void talos_corr_kernel(
    const float* __restrict__ y_hat, const float* __restrict__ y,
    const int* __restrict__ subject, float* __restrict__ acc) {
  __shared__ float yp[PAD_LEN];
  __shared__ float yh[T_LEN];

  const int b = blockIdx.x;
  const float* yr  = y     + (size_t)b * T_LEN;
  const float* yhr = y_hat + (size_t)b * T_LEN;
  const int tid = threadIdx.x;

  for (int i = tid; i < PAD_LEN; i += 64) {
    int src = i - HALF_K;
    yp[i] = (src >= 0 && src < T_LEN) ? yr[src] : 0.0f;
  }
  for (int i = tid; i < T_LEN; i += 64) yh[i] = yhr[i];
  __syncthreads();

  const int k = tid;  // lag (0..63; 61..63 are dead lanes, stay in-bounds)
  float corr = 0.0f, s2 = 0.0f, h = 0.0f;
#pragma unroll 8
  for (int t = 0; t < T_LEN; ++t) {
    float a = yh[t];       // uniform broadcast from LDS
    float v = yp[t + k];   // conflict-free: consecutive lanes -> consecutive banks
    corr = fmaf(a, v, corr);
    s2   = fmaf(v, v, s2);
    h    = fmaf(a, a, h);
  }

  if (k < KLAGS) {
    int s0 = subject[b] - 1;                        // subject in [1,64]
    atomicAdd(&acc[s0 * ACC_LD + k], s2 + h - 2.0f * corr);
  }
}

// ---------------------------------------------------------------------------
// Kernel 2: softmax over thea_s rows, then  total = sum_{s,k} acc[s,k]*P[s,k]
//         = trace(Acc * P^T), computed as 4 diagonal 16x16 tiles with
//         chained V_WMMA_F32_16X16X4_F32 (fp32, K in chunks of 4).
// Single wave32; EXEC is all-ones at every WMMA issue point.
//
// Operand layouts (ISA 7.12.2, 32-bit):
//   A 16x4 : VGPR0 = {lanes0-15: (M=lane,  K=0), lanes16-31: (M=lane-16, K=2)}
//            VGPR1 = K=1 / K=3
//   B 4x16 : symmetric: VGPR0 = {lanes0-15: (K=0, N=lane), lanes16-31: (K=2, N=lane-16)}
//   C 16x16: VGPR j = {lanes0-15: (M=j, N=lane), lanes16-31: (M=8+j, N=lane-16)}
//   => diagonal C[m,m]: lane j (j<8) holds C[j,j] in c[j];
//                       lane 24+j holds C[8+j,8+j] in c[j].
// ---------------------------------------------------------------------------
__global__ __launch_bounds__(32) void talos_finalize_kernel(
    const float* __restrict__ thea, const float* __restrict__ acc,
    float* __restrict__ out, float inv_scale) {
  __shared__ float P[S_SUBJ][ACC_LD];   // softmax probs, cols 61..63 zeroed
  __shared__ float A[S_SUBJ][ACC_LD];   // per-subject accumulated 240*mse

  const int lane = threadIdx.x;

  // Each lane owns 2 subject rows: softmax + copy of acc (uniform trip count,
  // no divergence -> EXEC stays all-ones).
  for (int r = lane; r < S_SUBJ; r += 32) {
    float mx = -INFINITY;
    for (int i = 0; i < KLAGS; ++i) mx = fmaxf(mx, thea[r * KLAGS + i]);
    float sum = 0.0f;
    for (int i = 0; i < KLAGS; ++i) {
      float e = __expf(thea[r * KLAGS + i] - mx);
      P[r][i] = e;
      sum += e;
    }
    float inv = 1.0f / sum;
    for (int i = 0; i < KLAGS; ++i) P[r][i] *= inv;
    for (int i = KLAGS; i < ACC_LD; ++i) P[r][i] = 0.0f;
    for (int i = 0; i < ACC_LD; ++i) A[r][i] = acc[r * ACC_LD + i];
  }
  __syncthreads();

  const int half = lane >> 4;   // 0: lanes 0-15, 1: lanes 16-31
  const int l16  = lane & 15;

  float diagsum = 0.0f;
  for (int d = 0; d < 4; ++d) {          // 4 diagonal 16x16 tiles
    v8f c = {0.f, 0.f, 0.f, 0.f, 0.f, 0.f, 0.f, 0.f};
#pragma unroll
    for (int kc = 0; kc < 16; ++kc) {    // contraction K=64 in chunks of 4
      const int kb = kc * 4 + 2 * half;
      v2f a, bb;
      // A_tile[m,k] = A[16d+m][k] ; per-lane: m=l16, k=kb(+1)
      a.x  = A[16 * d + l16][kb + 0];
      a.y  = A[16 * d + l16][kb + 1];
      // B_tile[k,n] = P[16d+n][k] ; per-lane: n=l16, k=kb(+1)
      bb.x = P[16 * d + l16][kb + 0];
      bb.y = P[16 * d + l16][kb + 1];
      c = __builtin_amdgcn_wmma_f32_16x16x4_f32(
          /*neg_a=*/false, a, /*neg_b=*/false, bb,
          /*c_mod=*/(short)0, c, /*reuse_a=*/false, /*reuse_b=*/false);
    }
    // Extract this tile's diagonal (reconverges before next tile's WMMAs).
    float dsum = 0.0f;
    if (lane < 8)        dsum = c[lane];
    else if (lane >= 24) dsum = c[lane - 24];
    diagsum += dsum;
  }

  // wave32 reduction
  for (int off = 16; off > 0; off >>= 1)
    diagsum += __shfl_xor(diagsum, off, 32);

  if (lane == 0) out[0] = diagsum * inv_scale;
}

// ---------------------------------------------------------------------------
// Inputs (setup_inputs order): y_hat (B,240) f32, y (B,240) f32,
// subject (B,) int, thea_s (64,61) f32.  Output: 1 f32 scalar.
// d_ws: 64x64 f32 accumulator (16 KB), zeroed each call (deterministic).
// ---------------------------------------------------------------------------
extern "C" void kernel_launch(void* const* d_in, const int* in_sizes, int n_in,
                              void* d_out, int out_size, void* d_ws, size_t ws_size,
                              hipStream_t stream) {
  const float* y_hat   = (const float*)d_in[0];
  const float* y       = (const float*)d_in[1];
  const int*   subject = (const int*)d_in[2];
  const float* thea    = (const float*)d_in[3];
  float*       out     = (float*)d_out;
  float*       acc     = (float*)d_ws;

  const int B = in_sizes[2];   // 16384

  hipMemsetAsync(acc, 0, S_SUBJ * ACC_LD * sizeof(float), stream);

  talos_corr_kernel<<<B, 64, 0, stream>>>(y_hat, y, subject, acc);

  const float inv_scale = 1.0f / (240.0f * (float)B);
  talos_finalize_kernel<<<1, 32, 0, stream>>>(thea, acc, out, inv_scale);
}